// PSNRLoss_27118423507430
// MI455X (gfx1250) — compile-verified
//
#include <hip/hip_runtime.h>
#include <math.h>

#define IMG_H 1080
#define IMG_W 1920
#define NFLOW 5
#define NPIX (IMG_H * IMG_W)
#define BLOCK 128                    // 1920 = 15 * 128 -> exact row tiling
#define GRIDX (IMG_W / BLOCK)        // 15 blocks per image row

// Native clang vector type (required by __builtin_nontemporal_load).
typedef float v2f __attribute__((ext_vector_type(2)));

// ---------------------------------------------------------------------------
// Kernel 1: warped-frame squared-error partial sums.
// grid = (15, 1080, NFLOW), block = 128  ->  h = blockIdx.y, no int division.
// Flow is streamed once -> non-temporal load (keeps the 50MB frame working
// set resident in the 192MB L2). Gathers use 32-bit element offsets so the
// backend can use SGPR-base + VGPR-offset addressing.
// Per-block SSE: wave32 shuffle tree + LDS, one global_atomic_add_f64.
// ---------------------------------------------------------------------------
__global__ __launch_bounds__(BLOCK) void warp_sse_kernel(
    const float* __restrict__ flow,    // [N, H, W, 2]
    const float* __restrict__ frame1,  // [3, H, W]
    const float* __restrict__ frame2,  // [3, H, W]
    double* __restrict__ sse)          // [N]
{
    const int n = blockIdx.z;
    const int h = blockIdx.y;
    const int w = blockIdx.x * BLOCK + threadIdx.x;
    const unsigned pix = (unsigned)h * IMG_W + (unsigned)w;   // < 2^21

    // One-shot 83MB stream: non-temporal.
    const v2f f = __builtin_nontemporal_load(
        (const v2f*)flow + ((unsigned)n * (unsigned)NPIX + pix));

    const float px = (float)w + f.x;
    const float py = (float)h + f.y;
    const float x0f = floorf(px);
    const float y0f = floorf(py);
    const float wx1 = px - x0f;
    const float wy1 = py - y0f;
    const float wx0 = 1.0f - wx1;
    const float wy0 = 1.0f - wy1;

    const int x0 = (int)x0f, y0 = (int)y0f;
    const int x1 = x0 + 1,   y1 = y0 + 1;

    const bool vx0 = (x0 >= 0) && (x0 <= IMG_W - 1);
    const bool vx1 = (x1 >= 0) && (x1 <= IMG_W - 1);
    const bool vy0 = (y0 >= 0) && (y0 <= IMG_H - 1);
    const bool vy1 = (y1 >= 0) && (y1 <= IMG_H - 1);

    const int xc0 = min(max(x0, 0), IMG_W - 1);
    const int xc1 = min(max(x1, 0), IMG_W - 1);
    const int yc0 = min(max(y0, 0), IMG_H - 1);
    const int yc1 = min(max(y1, 0), IMG_H - 1);

    // Validity folded into the bilinear weights (== reference's v*valid*w).
    const float w00 = wx0 * wy0 * (float)(vx0 && vy0);
    const float w10 = wx1 * wy0 * (float)(vx1 && vy0);
    const float w01 = wx0 * wy1 * (float)(vx0 && vy1);
    const float w11 = wx1 * wy1 * (float)(vx1 && vy1);

    // 32-bit element offsets (max 3*NPIX < 2^23).
    const unsigned o00 = (unsigned)yc0 * IMG_W + (unsigned)xc0;
    const unsigned o10 = (unsigned)yc0 * IMG_W + (unsigned)xc1;
    const unsigned o01 = (unsigned)yc1 * IMG_W + (unsigned)xc0;
    const unsigned o11 = (unsigned)yc1 * IMG_W + (unsigned)xc1;

    float acc = 0.0f;
    #pragma unroll
    for (int c = 0; c < 3; ++c) {
        const float* __restrict__ p1 = frame1 + (unsigned)c * (unsigned)NPIX;
        float v = p1[o00] * w00;
        v = fmaf(p1[o10], w10, v);
        v = fmaf(p1[o01], w01, v);
        v = fmaf(p1[o11], w11, v);
        v = truncf(v);
        const float d = frame2[(unsigned)c * (unsigned)NPIX + pix] - v;
        acc = fmaf(d, d, acc);
    }

    // wave32 intra-wave reduction (gfx1250 is wave32-only)
    #pragma unroll
    for (int off = 16; off > 0; off >>= 1)
        acc += __shfl_down(acc, off, 32);

    __shared__ float warp_part[BLOCK / 32];   // 4 waves
    const int lane = threadIdx.x & 31;
    const int wid  = threadIdx.x >> 5;
    if (lane == 0) warp_part[wid] = acc;
    __syncthreads();

    if (wid == 0) {
        float v = (lane < (BLOCK / 32)) ? warp_part[lane] : 0.0f;
        #pragma unroll
        for (int off = (BLOCK / 64); off > 0; off >>= 1)
            v += __shfl_down(v, off, 32);
        if (lane == 0)
            atomicAdd(&sse[n], (double)v);    // global_atomic_add_f64
    }
}

// ---------------------------------------------------------------------------
// Kernel 2: finalize. mse -> psnr -> weighted sum -> negate. Single scalar.
// ---------------------------------------------------------------------------
__global__ void finalize_kernel(const double* __restrict__ sse,
                                float* __restrict__ out)
{
    if (threadIdx.x == 0 && blockIdx.x == 0) {
        const double denom = 3.0 * (double)NPIX;
        double aggr = 0.0;
        for (int n = 0; n < NFLOW; ++n) {
            const double mse  = sse[n] / denom;
            const double psnr = 10.0 * log10((255.0 * 255.0) / mse);
            double wgt = 1.0;                       // 0.85^(NFLOW - n)
            for (int k = 0; k < NFLOW - n; ++k) wgt *= 0.85;
            aggr += psnr * wgt;
        }
        out[0] = (float)(-aggr);
    }
}

extern "C" void kernel_launch(void* const* d_in, const int* in_sizes, int n_in,
                              void* d_out, int out_size, void* d_ws, size_t ws_size,
                              hipStream_t stream)
{
    (void)in_sizes; (void)n_in; (void)out_size; (void)ws_size;

    const float* flow   = (const float*)d_in[0];  // [N,H,W,2] f32
    const float* frame1 = (const float*)d_in[1];  // [3,H,W]   f32
    const float* frame2 = (const float*)d_in[2];  // [3,H,W]   f32
    double*      sse    = (double*)d_ws;          // [N] accumulators
    float*       out    = (float*)d_out;          // scalar

    (void)hipMemsetAsync(sse, 0, NFLOW * sizeof(double), stream);

    dim3 block(BLOCK);
    dim3 grid(GRIDX, IMG_H, NFLOW);               // 15 x 1080 x 5
    warp_sse_kernel<<<grid, block, 0, stream>>>(flow, frame1, frame2, sse);
    finalize_kernel<<<1, 32, 0, stream>>>(sse, out);
}